// CombinedLoss_40157944218132
// MI455X (gfx1250) — compile-verified
//
#include <hip/hip_runtime.h>
#include <math.h>

#define DIM      512
#define NCLS     16
#define EPSV     1e-8f
#define MARGIN   0.2f
#define ALPHAV   0.1f
#define BETAV    1.05f

// float offsets into d_ws
#define OFF_ACC     0       // [8]: 1=intra_sum 2=inter_sum 3=n_terms
#define OFF_COUNTS  16      // [16]
#define OFF_SUMS    32      // [16*512]
#define OFF_CNORM   8224    // [16*512]  normalized centroids
#define OFF_PAIR    16416   // [256]     pair_f
#define OFF_CNT     16672   // [16]      close-pairs per class
#define OFF_TRIP    16688   // [n/8]     per-block triplet partials
#define ZERO_FLOATS 8224

typedef __attribute__((ext_vector_type(2))) float v2f;
typedef __attribute__((ext_vector_type(8))) float v8f;
typedef __attribute__((ext_vector_type(4))) unsigned int v4u;
typedef __attribute__((ext_vector_type(4))) int v4i;
typedef __attribute__((ext_vector_type(8))) int v8i;

#if defined(__has_builtin)
#if __has_builtin(__builtin_amdgcn_tensor_load_to_lds) && \
    __has_builtin(__builtin_amdgcn_s_wait_tensorcnt)
#define HAVE_TDM 1
#endif
#endif
#ifndef HAVE_TDM
#define HAVE_TDM 0
#endif

// ---------------- kernel 0: zero accumulators / counts / sums ----------------
__global__ void zero_ws_kernel(float* ws) {
    int i = blockIdx.x * blockDim.x + threadIdx.x;
    if (i < ZERO_FLOATS) ws[i] = 0.0f;
}

// ---------------- kernel 1: per-class sums + counts (LDS privatized) --------
__global__ void segsum_kernel(const float* __restrict__ emb,
                              const int* __restrict__ labels,
                              float* __restrict__ ws, int rows_per_block) {
    __shared__ float lsum[NCLS * DIM];   // 32 KB
    __shared__ float lcnt[NCLS];
    int t = threadIdx.x;                 // 256 threads
    for (int i = t; i < NCLS * DIM; i += 256) lsum[i] = 0.0f;
    if (t < NCLS) lcnt[t] = 0.0f;
    __syncthreads();

    int rbase = blockIdx.x * rows_per_block;
    for (int r = 0; r < rows_per_block; ++r) {
        int row = rbase + r;
        int lab = labels[row];                        // uniform broadcast load
        const float* e = emb + (size_t)row * DIM;
        lsum[lab * DIM + t]       += e[t];            // thread owns cols t,t+256
        lsum[lab * DIM + t + 256] += e[t + 256];
        if (t == 0) lcnt[lab] += 1.0f;
    }
    __syncthreads();

    float* sums   = ws + OFF_SUMS;
    float* counts = ws + OFF_COUNTS;
    for (int i = t; i < NCLS * DIM; i += 256) {
        float v = lsum[i];
        if (v != 0.0f) atomicAdd(&sums[i], v);
    }
    if (t < NCLS) atomicAdd(&counts[t], lcnt[t]);
}

// ---------------- kernel 2: centroids -> normalize -> pair_f, cnt -----------
__global__ void centroid_kernel(float* __restrict__ ws) {
    __shared__ float sred[256];
    int t = threadIdx.x;                 // 256 threads, single block
    float* counts = ws + OFF_COUNTS;
    float* sums   = ws + OFF_SUMS;
    float* cnorm  = ws + OFF_CNORM;
    float* pair   = ws + OFF_PAIR;
    float* cnt    = ws + OFF_CNT;

    for (int c = 0; c < NCLS; ++c) {
        float div = fmaxf(counts[c], 1.0f);
        float c0 = sums[c * DIM + t]       / div;
        float c1 = sums[c * DIM + t + 256] / div;
        sred[t] = c0 * c0 + c1 * c1;
        __syncthreads();
        for (int s = 128; s > 0; s >>= 1) {
            if (t < s) sred[t] += sred[t + s];
            __syncthreads();
        }
        float nrm = fmaxf(sqrtf(sred[0]), EPSV);
        cnorm[c * DIM + t]       = c0 / nrm;
        cnorm[c * DIM + t + 256] = c1 / nrm;
        __syncthreads();
    }

    // pairwise centroid cosine distances: thread t -> (i,j)
    int i = t >> 4, j = t & 15;
    float dot = 0.0f;
    for (int k = 0; k < DIM; ++k)
        dot = fmaf(cnorm[i * DIM + k], cnorm[j * DIM + k], dot);
    float cd = 1.0f - dot;
    bool ok = (cd <= BETAV) && (i != j) && (counts[i] > 0.0f) && (counts[j] > 0.0f);
    pair[t] = ok ? 1.0f : 0.0f;
    __syncthreads();
    if (t < NCLS) {
        float s = 0.0f;
        for (int jj = 0; jj < NCLS; ++jj) s += pair[t * NCLS + jj];
        cnt[t] = s;
    }
}

// ---------------- kernel 3: triplet loss, one wave32 per row ----------------
__device__ __forceinline__ float dot4(float4 a, float4 b) {
    return fmaf(a.x, b.x, fmaf(a.y, b.y, fmaf(a.z, b.z, a.w * b.w)));
}

__global__ void triplet_kernel(const float* __restrict__ A,
                               const float* __restrict__ P,
                               const float* __restrict__ Ng,
                               float* __restrict__ trip_part) {
    __shared__ float wsum[8];
    int lane = threadIdx.x & 31;
    int wave = threadIdx.x >> 5;
    int row  = blockIdx.x * 8 + wave;
    const float4* a4 = (const float4*)(A  + (size_t)row * DIM);
    const float4* p4 = (const float4*)(P  + (size_t)row * DIM);
    const float4* n4 = (const float4*)(Ng + (size_t)row * DIM);

    float aa = 0.f, pp = 0.f, nn = 0.f, ap = 0.f, an = 0.f;
#pragma unroll
    for (int i = 0; i < 4; ++i) {
        int idx = lane + i * 32;
        float4 a = a4[idx], p = p4[idx], g = n4[idx];
        aa += dot4(a, a); pp += dot4(p, p); nn += dot4(g, g);
        ap += dot4(a, p); an += dot4(a, g);
    }
    for (int off = 16; off > 0; off >>= 1) {
        aa += __shfl_down(aa, off); pp += __shfl_down(pp, off);
        nn += __shfl_down(nn, off); ap += __shfl_down(ap, off);
        an += __shfl_down(an, off);
    }
    if (lane == 0) {
        float da = fmaxf(sqrtf(aa), EPSV);
        float dp = fmaxf(sqrtf(pp), EPSV);
        float dn = fmaxf(sqrtf(nn), EPSV);
        float pd = 1.0f - ap / (da * dp);
        float nd = 1.0f - an / (da * dn);
        wsum[wave] = fmaxf(pd - nd + MARGIN, 0.0f);
    }
    __syncthreads();
    if (threadIdx.x == 0) {
        float s = 0.f;
        for (int w = 0; w < 8; ++w) s += wsum[w];
        trip_part[blockIdx.x] = s;
    }
}

// ---------------- kernel 4: ATN via WMMA f32 16x16x4, TDM-fed tiles ---------
// One wave per 16-row tile: D_tile[16x16] = emb_tile[16x512] * cnorm^T[512x16]
#define ATN_WAVES 2
#define CH_K      128                   // k-chunk (DWORDs per row per chunk)
#define ROWSTR    132                   // 128 + 4 pad dwords -> stride%64 == 4
#define TILEBUF   (16 * ROWSTR)         // 2112 dwords = 8448 B per buffer

#if HAVE_TDM
// Build D# and issue TENSOR_LOAD_TO_LDS for a 16 x CH_K f32 tile, row stride
// 512 elems in memory, LDS padding 4 dwords per 128-dword row (bank spread).
__device__ __forceinline__ void tdm_load_tile(unsigned lds_off, const float* gptr) {
    unsigned long long ga = (unsigned long long)(size_t)gptr;
    v4u g0;
    g0.x = 1u;                                          // count=1, user D#
    g0.y = lds_off;                                     // lds_addr (bytes)
    g0.z = (unsigned)(ga & 0xFFFFFFFFu);                // global_addr[31:0]
    g0.w = (unsigned)((ga >> 32) & 0x1FFFFFFu) | (2u << 30); // addr[56:32]|type=2
    v8i g1;
    g1[0] = (int)((2u << 16)     // data_size = 4 B
                | (1u << 20)     // pad_enable
                | (6u << 22)     // pad_interval code 6 = 128 dwords
                | (3u << 25));   // pad_amount  code 3 = 4 dwords
    g1[1] = (int)(512u << 16);   // abar_addr=0 | tensor_dim0[15:0]=512
    g1[2] = (int)(16u << 16);    // tensor_dim0[31:16]=0 | tensor_dim1[15:0]=16
    g1[3] = (int)(128u << 16);   // tensor_dim1[31:16]=0 | tile_dim0=128
    g1[4] = 16;                  // tile_dim1=16 | tile_dim2=0
    g1[5] = 512;                 // tensor_dim0_stride[31:0]=512 elems
    g1[6] = 0;
    g1[7] = 0;
    v4i z4 = {0, 0, 0, 0};
#if __clang_major__ >= 23
    v8i z8 = {0, 0, 0, 0, 0, 0, 0, 0};
    __builtin_amdgcn_tensor_load_to_lds(g0, g1, z4, z4, z8, 0);
#else
    __builtin_amdgcn_tensor_load_to_lds(g0, g1, z4, z4, 0);
#endif
}
#endif

__global__ void atn_kernel(const float* __restrict__ emb,
                           const int* __restrict__ labels,
                           float* __restrict__ ws) {
#if HAVE_TDM
    __shared__ __align__(16) float abuf[ATN_WAVES][2][TILEBUF];
#endif
    __shared__ float wred[3][ATN_WAVES];
    const float* cnorm = ws + OFF_CNORM;
    const float* pair  = ws + OFF_PAIR;
    const float* cnt   = ws + OFF_CNT;
    float* acc = ws + OFF_ACC;

    int lane  = threadIdx.x & 31;
    int wave  = threadIdx.x >> 5;
    int tile  = blockIdx.x * ATN_WAVES + wave;
    int rbase = tile * 16;
    int mlo   = lane & 15;            // A: row index; B/D: column index
    int koff  = (lane >> 4) << 1;     // 0 (lanes 0-15) or 2 (lanes 16-31)

    const float* crow = cnorm + (size_t)mlo * DIM + koff;

    v8f acct = {0.f, 0.f, 0.f, 0.f, 0.f, 0.f, 0.f, 0.f};
    float ss = 0.0f;                  // partial sum-of-squares of this lane's row

#if HAVE_TDM
    const float* gtile = emb + (size_t)rbase * DIM;
    // double-buffered: DMA chunk c+1 while WMMA consumes chunk c
    tdm_load_tile((unsigned)(size_t)(const void*)&abuf[wave][0][0], gtile);
#pragma unroll
    for (int c = 0; c < 4; ++c) {
        if (c < 3) {
            tdm_load_tile((unsigned)(size_t)(const void*)&abuf[wave][(c + 1) & 1][0],
                          gtile + (c + 1) * CH_K);
            __builtin_amdgcn_s_wait_tensorcnt((short)1);   // chunk c landed
        } else {
            __builtin_amdgcn_s_wait_tensorcnt((short)0);
        }
        const float* lrow = &abuf[wave][c & 1][0] + mlo * ROWSTR + koff;
        const float* cro2 = crow + c * CH_K;
#pragma unroll 4
        for (int k = 0; k < CH_K; k += 4) {
            float2 av = *(const float2*)(lrow + k);     // ds_load_b64, bank-clean
            float2 bv = *(const float2*)(cro2 + k);
            ss = fmaf(av.x, av.x, fmaf(av.y, av.y, ss));
            v2f Af = {av.x, av.y};
            v2f Bf = {bv.x, bv.y};
            acct = __builtin_amdgcn_wmma_f32_16x16x4_f32(
                false, Af, false, Bf, (short)0, acct, false, false);
        }
    }
#else
    const float* erow = emb + (size_t)(rbase + mlo) * DIM + koff;
#pragma unroll 4
    for (int k = 0; k < DIM; k += 4) {
        float2 av = *(const float2*)(erow + k);
        float2 bv = *(const float2*)(crow + k);
        ss = fmaf(av.x, av.x, fmaf(av.y, av.y, ss));
        v2f Af = {av.x, av.y};
        v2f Bf = {bv.x, bv.y};
        acct = __builtin_amdgcn_wmma_f32_16x16x4_f32(
            false, Af, false, Bf, (short)0, acct, false, false);
    }
#endif

    // combine the two K-half partial norms: lane l then holds ||row (l&15)||^2
    ss += __shfl_xor(ss, 16);

    float intra = 0.f, inter = 0.f, nterm = 0.f;
    int n = mlo;
    int mbase = (lane >> 4) * 8;      // D layout: lanes 16-31 hold M = j+8
#pragma unroll
    for (int j = 0; j < 8; ++j) {
        int m = mbase + j;
        float rn  = fmaxf(sqrtf(__shfl(ss, m)), EPSV);
        float d   = 1.0f - acct[j] / rn;          // cos distance D[m][n]
        int   lab = labels[rbase + m];            // uniform per half-wave
        float pf  = pair[lab * NCLS + n];
        inter += pf * fmaxf(BETAV - d, 0.0f);
        if (n == lab) {                           // exactly once per row
            float wv = cnt[lab];
            intra += wv * fmaxf(d - ALPHAV, 0.0f);
            nterm += wv;
        }
    }
    for (int off = 16; off > 0; off >>= 1) {
        intra += __shfl_down(intra, off);
        inter += __shfl_down(inter, off);
        nterm += __shfl_down(nterm, off);
    }
    if (lane == 0) { wred[0][wave] = intra; wred[1][wave] = inter; wred[2][wave] = nterm; }
    __syncthreads();
    if (threadIdx.x == 0) {
        float a = 0.f, b = 0.f, c = 0.f;
        for (int w = 0; w < ATN_WAVES; ++w) {
            a += wred[0][w]; b += wred[1][w]; c += wred[2][w];
        }
        atomicAdd(&acc[1], a);
        atomicAdd(&acc[2], b);
        atomicAdd(&acc[3], c);
    }
}

// ---------------- kernel 5: final combine -----------------------------------
__global__ void final_kernel(const float* __restrict__ ws, float* __restrict__ out,
                             int ntrip, float invN) {
    __shared__ float sred[256];
    int t = threadIdx.x;
    float s = 0.f;
    for (int i = t; i < ntrip; i += 256) s += ws[OFF_TRIP + i];
    sred[t] = s;
    __syncthreads();
    for (int k = 128; k > 0; k >>= 1) {
        if (t < k) sred[t] += sred[t + k];
        __syncthreads();
    }
    if (t == 0) {
        float trip  = sred[0] * invN;
        float intra = ws[OFF_ACC + 1];
        float inter = ws[OFF_ACC + 2];
        float nterm = ws[OFF_ACC + 3];
        float safe  = fmaxf(nterm, 1.0f);
        float atn   = (nterm > 0.0f) ? (intra / safe + inter / safe) : 0.0f;
        out[0] = trip + atn;
    }
}

extern "C" void kernel_launch(void* const* d_in, const int* in_sizes, int n_in,
                              void* d_out, int out_size, void* d_ws, size_t ws_size,
                              hipStream_t stream) {
    const float* anchor   = (const float*)d_in[0];
    const float* positive = (const float*)d_in[1];
    const float* negative = (const float*)d_in[2];
    const float* emb      = (const float*)d_in[3];
    const int*   labels   = (const int*)d_in[4];
    float* ws  = (float*)d_ws;
    float* out = (float*)d_out;

    int n = in_sizes[4];              // N = 65536
    int zblocks = (ZERO_FLOATS + 255) / 256;
    zero_ws_kernel<<<zblocks, 256, 0, stream>>>(ws);

    int segblocks = 256;
    int rows_per_block = n / segblocks;
    segsum_kernel<<<segblocks, 256, 0, stream>>>(emb, labels, ws, rows_per_block);

    centroid_kernel<<<1, 256, 0, stream>>>(ws);

    int tripblocks = n / 8;           // 8 waves per block, one row per wave
    triplet_kernel<<<tripblocks, 256, 0, stream>>>(anchor, positive, negative,
                                                   ws + OFF_TRIP);

    int atnblocks = n / (16 * ATN_WAVES);   // one 16-row tile per wave
    atn_kernel<<<atnblocks, ATN_WAVES * 32, 0, stream>>>(emb, labels, ws);

    final_kernel<<<1, 256, 0, stream>>>(ws, out, tripblocks, 1.0f / (float)n);
}